// AmbientDrone_14585708937547
// MI455X (gfx1250) — compile-verified
//
#include <hip/hip_runtime.h>
#include <hip/hip_bf16.h>
#include <math.h>

// ---- problem constants -----------------------------------------------------
#define SRATE   44100.0f
#define NSAMP   88200
#define NVOICE  64
#define NCHUNK  5513              // ceil(88200/16)
#define NTILE   345               // ceil(5513/16)
#define FPAD    (NTILE * 256)     // 88320 padded samples per voice (345*16 chunks)
#define CB      80                // floats of per-voice constants

typedef __attribute__((ext_vector_type(2))) float v2f;
typedef __attribute__((ext_vector_type(8))) float v8f;

__constant__ float PLOW[14]  = {20.f,50.f,0.f,0.f,0.f,0.f,0.1f,0.f,0.01f,0.1f,0.f,0.1f,100.f,0.5f};
__constant__ float PHIGH[14] = {200.f,800.f,1.f,1.f,1.f,0.3f,5.f,50.f,2.f,2.f,1.f,3.f,8000.f,10.f};

__device__ __forceinline__ float clamp01(float x){ return fminf(fmaxf(x, 0.f), 1.f); }

// ---- kernel 0: per-voice constants ----------------------------------------
// cb layout: [0]=2pi*f1 [1]=2pi*f2 [2]=shape [3..5]=l1,l2,ln [6]=2pi*lfo_rate
// [7]=lfo_depth/1200 [8..11]=ADSR [12]=lvl(NOTE_ON) [13]=pi*k/2
// [14..18]=b0,b1,b2,a1,a2 [19..35]=h[0..16] [36..51]=p_t=h[t+1] [52..67]=q_t=-a2*h[t]
__global__ void k_prep(const float* __restrict__ params, float* __restrict__ cbuf) {
  int v = threadIdx.x;
  if (v >= NVOICE) return;
  float p[14];
  for (int i = 0; i < 14; ++i) p[i] = params[v*14 + i] * (PHIGH[i] - PLOW[i]) + PLOW[i];
  float* cb = cbuf + v*CB;
  const float twopi = 6.28318530717958647692f;
  float f2 = p[1];
  cb[0] = twopi * p[0];
  cb[1] = twopi * f2;
  cb[2] = p[2];
  cb[3] = p[3]; cb[4] = p[4]; cb[5] = p[5];
  cb[6] = twopi * p[6];
  cb[7] = p[7] * (1.f/1200.f);
  float attack = p[8], decay = p[9], sustain = p[10], release = p[11];
  cb[8] = attack; cb[9] = decay; cb[10] = sustain; cb[11] = release;
  cb[12] = (1.0f <= attack) ? (1.0f / attack)
         : (sustain + (1.f - sustain) * clamp01(1.f - (1.f - attack)/decay));
  float k = 12000.f / (f2 * log10f(f2));
  cb[13] = 0.5f * 3.14159265358979323846f * k;
  // biquad lowpass coefficients
  float fc = fminf(fmaxf(p[12], 20.f), SRATE*0.5f - 100.f);
  float q  = fminf(fmaxf(p[13], 0.1f), 20.f);
  float w0 = twopi * fc / SRATE;
  float alpha = sinf(w0) / (2.f*q);
  float cw = cosf(w0);
  float a0 = 1.f + alpha;
  float b0 = (1.f - cw) * 0.5f / a0;
  float b1 = (1.f - cw) / a0;
  float a1 = -2.f * cw / a0;
  float a2 = (1.f - alpha) / a0;
  cb[14] = b0; cb[15] = b1; cb[16] = b0; cb[17] = a1; cb[18] = a2;
  // AR impulse response, block-scan correction vectors
  float h[17];
  h[0] = 1.f; h[1] = -a1;
  for (int i = 2; i < 17; ++i) h[i] = -a1*h[i-1] - a2*h[i-2];
  for (int i = 0; i < 17; ++i) cb[19 + i] = h[i];
  for (int t = 0; t < 16; ++t) cb[36 + t] = h[t + 1];    // multiplies y[-1]
  for (int t = 0; t < 16; ++t) cb[52 + t] = -a2 * h[t];  // multiplies y[-2]
}

// ---- kernel 1: signal generation + fused FIR ------------------------------
__device__ float sample_x(const float* __restrict__ cb, const float* __restrict__ nz, int t) {
  if (t < 0 || t >= NSAMP) return 0.f;
  float tt = (float)t * (1.0f / SRATE);
  float attack = cb[8], decay = cb[9], sustain = cb[10], release = cb[11];
  float amp_on = (tt <= attack) ? (tt / attack)
               : (sustain + (1.f - sustain) * clamp01(1.f - (tt - attack)/decay));
  float rel = cb[12] * clamp01(1.f - (tt - 1.f)/release);
  float env = (tt < 1.f) ? amp_on : rel;
  float vco1 = sinf(cb[0] * tt);
  float ph2  = cb[1] * tt;
  float sq   = tanhf(cb[13] * sinf(ph2));
  float shape = cb[2];
  float vco2 = (1.f - 0.5f*shape) * sq * (1.f + shape * cosf(0.5f*ph2));
  float lfo  = exp2f(sinf(cb[6]*tt) * cb[7]);
  float mixed = cb[3]*vco1 + cb[4]*vco2*lfo + cb[5]*nz[t];
  return mixed * env;
}

__global__ void k_gen(const float* __restrict__ noise, const float* __restrict__ cbuf,
                      float* __restrict__ fbuf) {
  int v = blockIdx.x / NTILE;
  int t = (blockIdx.x % NTILE) * 256 + threadIdx.x;   // t in [0, FPAD)
  const float* cb = cbuf + v*CB;
  const float* nz = noise + v*NSAMP;
  float f = 0.f;
  if (t < NSAMP) {
    float x0 = sample_x(cb, nz, t);
    float x1 = sample_x(cb, nz, t - 1);
    float x2 = sample_x(cb, nz, t - 2);
    f = cb[14]*x0 + cb[15]*x1 + cb[16]*x2;
  }
  fbuf[v*FPAD + t] = f;
}

// ---- kernel 2: WMMA block-scan bulk: Z = H * F ----------------------------
// One wave per voice. A = 16x16 lower-triangular Toeplitz(h), split into four
// 16x4 K-tiles. B tile = 4x16 (K x chunks). F tiles (1KB each) are staged into
// LDS with double-buffered GLOBAL_LOAD_ASYNC_TO_LDS_B128 (ASYNCcnt-tracked)
// so the next tile's DMA overlaps the current tile's WMMA chain.
__global__ void __launch_bounds__(32) k_wmma(const float* __restrict__ cbuf,
                                             const float* __restrict__ fbuf,
                                             float* __restrict__ zbuf) {
  __shared__ float lds[2][256];     // two 1KB tile buffers
  int v    = blockIdx.x;
  int lane = threadIdx.x;
  int hi   = lane >> 4;     // lane half selects K pair / row half
  int lm   = lane & 15;     // M for A, N (chunk column) for B/D
  const float* cb = cbuf + v*CB;
  float h[17];
#pragma unroll
  for (int i = 0; i < 17; ++i) h[i] = cb[19 + i];
  // A-matrix 16x4 layout: VGPR e holds K = 4*kt + e + 2*hi, row M = lm
  v2f A[4];
#pragma unroll
  for (int kt = 0; kt < 4; ++kt) {
    int j0 = 4*kt + 2*hi;
    A[kt].x = (lm >= j0    ) ? h[lm - j0    ] : 0.f;
    A[kt].y = (lm >= j0 + 1) ? h[lm - j0 - 1] : 0.f;
  }
  const float* fv = fbuf + v*FPAD;
  float*       zv = zbuf + v*FPAD;

  // per-lane LDS/global addresses for the async tile copy (32 lanes x 16B x 2 = 1KB)
  unsigned lds_off[2];
  lds_off[0] = (unsigned)(unsigned long long)(uintptr_t)&lds[0][lane*4];
  lds_off[1] = (unsigned)(unsigned long long)(uintptr_t)&lds[1][lane*4];

  // issue async DMA of 1KB tile 'tile' into LDS buffer 'b'
  auto issue = [&](int b, int tile) {
    unsigned long long g = (unsigned long long)(uintptr_t)(fv + tile*256 + lane*4);
    asm volatile("global_load_async_to_lds_b128 %0, %1, off\n\t"
                 "global_load_async_to_lds_b128 %0, %1, off offset:512"
                 :: "v"(lds_off[b]), "v"(g) : "memory");
  };

  issue(0, 0);                      // prologue: tile 0 -> buf 0
  for (int tile = 0; tile < NTILE; ++tile) {
    int nxt = (tile + 1 < NTILE) ? tile + 1 : NTILE - 1;
    issue((tile + 1) & 1, nxt);     // prefetch next tile into other buffer
    // async loads complete in order: <=2 outstanding means tile's buffer landed
    asm volatile("s_wait_asynccnt 0x2" ::: "memory");
    const float* lp = &lds[tile & 1][0];
    v8f acc = (v8f){0.f,0.f,0.f,0.f,0.f,0.f,0.f,0.f};
#pragma unroll
    for (int kt = 0; kt < 4; ++kt) {
      int j0 = 4*kt + 2*hi;
      float2 bb = *(const float2*)&lp[lm*16 + j0];   // ds_load_b64 from staged tile
      v2f B; B.x = bb.x; B.y = bb.y;
      // (neg_a, A, neg_b, B, c_mod, C, reuse_a, reuse_b)
      acc = __builtin_amdgcn_wmma_f32_16x16x4_f32(false, A[kt], false, B,
                                                  (short)0, acc, false, false);
    }
    // D layout: VGPR r holds rows r (lanes 0-15) and r+8 (lanes 16-31), col = lm
    int base = (tile*16 + lm) * 16;   // sample index of this lane's chunk start
    float* zp = zv + base + 8*hi;     // 8 contiguous samples
    float4 s0 = make_float4(acc[0], acc[1], acc[2], acc[3]);
    float4 s1 = make_float4(acc[4], acc[5], acc[6], acc[7]);
    *(float4*)(zp)     = s0;
    *(float4*)(zp + 4) = s1;
  }
}

// ---- kernel 3: serial chunk-state propagation (64 voices in parallel) -----
__global__ void k_scan(const float* __restrict__ cbuf, const float* __restrict__ zbuf,
                       float2* __restrict__ sbuf) {
  int v = blockIdx.x * blockDim.x + threadIdx.x;
  if (v >= NVOICE) return;
  const float* cb = cbuf + v*CB;
  float p14 = cb[36 + 14], p15 = cb[36 + 15];
  float q14 = cb[52 + 14], q15 = cb[52 + 15];
  const float* zv = zbuf + v*FPAD;
  float2* S = sbuf + v*NCHUNK;
  float s1 = 0.f, s2 = 0.f;   // y[-1], y[-2] entering the chunk
  for (int c = 0; c < NCHUNK; ++c) {
    __builtin_prefetch(&zv[(c + 8)*16 + 14], 0, 0);   // global_prefetch_b8
    S[c] = make_float2(s1, s2);
    float z14 = zv[c*16 + 14];
    float z15 = zv[c*16 + 15];
    float y14 = z14 + p14*s1 + q14*s2;
    float y15 = z15 + p15*s1 + q15*s2;
    s1 = y15; s2 = y14;
  }
}

// ---- kernel 4: apply state corrections, y overwrites z in place -----------
__global__ void k_correct(const float* __restrict__ cbuf, const float2* __restrict__ sbuf,
                          float* __restrict__ zbuf) {
  int v = blockIdx.x / NTILE;
  int r = (blockIdx.x % NTILE) * 256 + threadIdx.x;
  if (r >= NCHUNK * 16) return;
  int c = r >> 4, t = r & 15;
  const float* cb = cbuf + v*CB;
  float2 s = sbuf[v*NCHUNK + c];
  float y = zbuf[v*FPAD + r] + cb[36 + t]*s.x + cb[52 + t]*s.y;
  zbuf[v*FPAD + r] = y;
}

// ---- kernel 5: per-voice peak ---------------------------------------------
__global__ void k_reduce(const float* __restrict__ zbuf, float* __restrict__ mbuf) {
  __shared__ float sm[256];
  int v = blockIdx.x;
  const float* zv = zbuf + v*FPAD;
  float m = 0.f;
  for (int t = threadIdx.x; t < NSAMP; t += 256) m = fmaxf(m, fabsf(zv[t]));
  sm[threadIdx.x] = m;
  __syncthreads();
  for (int s = 128; s > 0; s >>= 1) {
    if (threadIdx.x < s) sm[threadIdx.x] = fmaxf(sm[threadIdx.x], sm[threadIdx.x + s]);
    __syncthreads();
  }
  if (threadIdx.x == 0) mbuf[v] = fmaxf(sm[0], 1e-8f);
}

// ---- kernel 6: normalize to output ----------------------------------------
__global__ void k_norm(const float* __restrict__ zbuf, const float* __restrict__ mbuf,
                       float* __restrict__ out) {
  int v = blockIdx.x / NTILE;
  int t = (blockIdx.x % NTILE) * 256 + threadIdx.x;
  if (t >= NSAMP) return;
  out[v*NSAMP + t] = zbuf[v*FPAD + t] * (0.9f / mbuf[v]);
}

// ---- host entry ------------------------------------------------------------
extern "C" void kernel_launch(void* const* d_in, const int* in_sizes, int n_in,
                              void* d_out, int out_size, void* d_ws, size_t ws_size,
                              hipStream_t stream) {
  (void)in_sizes; (void)n_in; (void)out_size; (void)ws_size;
  const float* params = (const float*)d_in[0];   // (64, 14) f32
  const float* noise  = (const float*)d_in[1];   // (64, 88200) f32

  // workspace carve-up (floats): cbuf | fbuf | zbuf | sbuf | mbuf  (~46 MiB)
  float*  ws   = (float*)d_ws;
  float*  cbuf = ws;                                   // 64*80
  float*  fbuf = ws + NVOICE*CB;                       // 64*FPAD
  float*  zbuf = fbuf + (size_t)NVOICE*FPAD;           // 64*FPAD
  float2* sbuf = (float2*)(zbuf + (size_t)NVOICE*FPAD);// 64*NCHUNK float2
  float*  mbuf = (float*)(sbuf + (size_t)NVOICE*NCHUNK);

  k_prep   <<<1, 64, 0, stream>>>(params, cbuf);
  k_gen    <<<NVOICE*NTILE, 256, 0, stream>>>(noise, cbuf, fbuf);
  k_wmma   <<<NVOICE, 32, 0, stream>>>(cbuf, fbuf, zbuf);
  k_scan   <<<2, 32, 0, stream>>>(cbuf, zbuf, sbuf);
  k_correct<<<NVOICE*NTILE, 256, 0, stream>>>(cbuf, sbuf, zbuf);
  k_reduce <<<NVOICE, 256, 0, stream>>>(zbuf, mbuf);
  k_norm   <<<NVOICE*NTILE, 256, 0, stream>>>(zbuf, mbuf, (float*)d_out);
}